// DeepGraphSAGE_41248865911073
// MI455X (gfx1250) — compile-verified
//
#include <hip/hip_runtime.h>
#include <hip/hip_bf16.h>

typedef __attribute__((ext_vector_type(2))) float v2f;
typedef __attribute__((ext_vector_type(8))) float v8f;

#define F 128  // feature width for layers 1/2 (F_IN == HID == 128)

// ---------------------------------------------------------------------------
// utility kernels
// ---------------------------------------------------------------------------
__global__ __launch_bounds__(256) void k_zero(float* __restrict__ p, size_t cnt) {
  size_t i = (size_t)blockIdx.x * blockDim.x + threadIdx.x;
  if (i < cnt) p[i] = 0.0f;
}

__global__ __launch_bounds__(256) void k_degree(const int* __restrict__ dst,
                                                float* __restrict__ deg, int nE) {
  int e = blockIdx.x * blockDim.x + threadIdx.x;
  if (e < nE) atomicAdd(&deg[dst[e]], 1.0f);
}

__global__ __launch_bounds__(256) void k_deg_inv(float* __restrict__ deg, int n) {
  int i = blockIdx.x * blockDim.x + threadIdx.x;
  if (i < n) {
    float d = deg[i];
    deg[i] = d > 0.0f ? 1.0f / d : 0.0f;
  }
}

// ---------------------------------------------------------------------------
// edge scatter-add: one wave32 per edge, 32 lanes x float4 cover a 128-f row
// ---------------------------------------------------------------------------
__global__ __launch_bounds__(256) void k_scatter(const int* __restrict__ src,
                                                 const int* __restrict__ dst,
                                                 const float* __restrict__ feat,
                                                 float* __restrict__ agg, int nE) {
  long long tid = (long long)blockIdx.x * blockDim.x + threadIdx.x;
  int e = (int)(tid >> 5);
  if (e >= nE) return;
  int fo = ((int)tid & 31) * 4;
  const float4 v = *reinterpret_cast<const float4*>(feat + (size_t)src[e] * F + fo);
  float* d = agg + (size_t)dst[e] * F + fo;
  atomicAdd(d + 0, v.x);
  atomicAdd(d + 1, v.y);
  atomicAdd(d + 2, v.z);
  atomicAdd(d + 3, v.w);
}

// ---------------------------------------------------------------------------
// SAGE dense update, OUT = 128:
//   H[i,:] = relu( (agg[i,:] * deg_inv[i]) @ Wl^T + bl + X[i,:] @ Wr^T )
// One wave per 16-row tile, 8 column tiles of 16, K=128 via 32 chained
// V_WMMA_F32_16X16X4_F32 per operand matrix. fp32 end to end.
// ---------------------------------------------------------------------------
__global__ __launch_bounds__(256) void k_gemm128(
    const float* __restrict__ agg, const float* __restrict__ xin,
    const float* __restrict__ Wl, const float* __restrict__ bl,
    const float* __restrict__ Wr, const float* __restrict__ deg_inv,
    float* __restrict__ hout, int n, int do_relu) {
  const int K = F;
  int lane = threadIdx.x & 31;
  int wave = threadIdx.x >> 5;
  int row0 = (blockIdx.x * 8 + wave) * 16;   // uniform per wave
  if (row0 >= n) return;                     // whole-wave exit, EXEC stays full

  int mloc = lane & 15;                      // A row within tile; also B column
  int koff = (lane >> 4) * 2;                // K sub-offset per ISA f32 layout
  int m = row0 + mloc;
  int mc = m < n ? m : n - 1;                // clamp tail-row loads
  float dscale = deg_inv[mc];
  const float* arow = agg + (size_t)mc * K;
  const float* xrow = xin + (size_t)mc * K;

  for (int ct = 0; ct < 8; ++ct) {
    int ncol = ct * 16 + mloc;
    float bias = bl[ncol];
    v8f c;
#pragma unroll
    for (int i = 0; i < 8; ++i) c[i] = bias;  // bias fused into accumulator

    const float* wlrow = Wl + (size_t)ncol * K;
    const float* wrrow = Wr + (size_t)ncol * K;

#pragma unroll
    for (int kt = 0; kt < 32; ++kt) {         // agg @ Wl^T (mean-scaled)
      int k0 = kt * 4 + koff;
      v2f a, b;
      a[0] = arow[k0] * dscale; a[1] = arow[k0 + 1] * dscale;
      b[0] = wlrow[k0];         b[1] = wlrow[k0 + 1];
      c = __builtin_amdgcn_wmma_f32_16x16x4_f32(false, a, false, b, (short)0, c,
                                                false, false);
    }
#pragma unroll
    for (int kt = 0; kt < 32; ++kt) {         // x @ Wr^T
      int k0 = kt * 4 + koff;
      v2f a, b;
      a[0] = xrow[k0]; a[1] = xrow[k0 + 1];
      b[0] = wrrow[k0]; b[1] = wrrow[k0 + 1];
      c = __builtin_amdgcn_wmma_f32_16x16x4_f32(false, a, false, b, (short)0, c,
                                                false, false);
    }

    int rbase = (lane >> 4) * 8;              // D layout: vgpr r -> M = r + 8*(lane>=16)
#pragma unroll
    for (int r = 0; r < 8; ++r) {
      int mr = row0 + rbase + r;
      if (mr < n) {
        float v = c[r];
        if (do_relu) v = v > 0.0f ? v : 0.0f;
        hout[(size_t)mr * F + ncol] = v;
      }
    }
  }
}

// ---------------------------------------------------------------------------
// Final layer, OUT = 7 (single 16-wide column tile, cols >= 7 masked)
// ---------------------------------------------------------------------------
__global__ __launch_bounds__(256) void k_gemm_out7(
    const float* __restrict__ agg, const float* __restrict__ xin,
    const float* __restrict__ Wl, const float* __restrict__ bl,
    const float* __restrict__ Wr, const float* __restrict__ deg_inv,
    float* __restrict__ logits, int n) {
  const int K = F;
  int lane = threadIdx.x & 31;
  int wave = threadIdx.x >> 5;
  int row0 = (blockIdx.x * 8 + wave) * 16;
  if (row0 >= n) return;

  int mloc = lane & 15;
  int koff = (lane >> 4) * 2;
  int m = row0 + mloc;
  int mc = m < n ? m : n - 1;
  float dscale = deg_inv[mc];
  const float* arow = agg + (size_t)mc * K;
  const float* xrow = xin + (size_t)mc * K;

  int ncol = mloc;
  bool cvalid = ncol < 7;
  float wm = cvalid ? 1.0f : 0.0f;            // zero-mask invalid B columns
  const float* wlrow = Wl + (size_t)(cvalid ? ncol : 0) * K;
  const float* wrrow = Wr + (size_t)(cvalid ? ncol : 0) * K;

  float bias = cvalid ? bl[ncol] : 0.0f;
  v8f c;
#pragma unroll
  for (int i = 0; i < 8; ++i) c[i] = bias;

#pragma unroll
  for (int kt = 0; kt < 32; ++kt) {
    int k0 = kt * 4 + koff;
    v2f a, b;
    a[0] = arow[k0] * dscale; a[1] = arow[k0 + 1] * dscale;
    b[0] = wlrow[k0] * wm;    b[1] = wlrow[k0 + 1] * wm;
    c = __builtin_amdgcn_wmma_f32_16x16x4_f32(false, a, false, b, (short)0, c,
                                              false, false);
  }
#pragma unroll
  for (int kt = 0; kt < 32; ++kt) {
    int k0 = kt * 4 + koff;
    v2f a, b;
    a[0] = xrow[k0];       a[1] = xrow[k0 + 1];
    b[0] = wrrow[k0] * wm; b[1] = wrrow[k0 + 1] * wm;
    c = __builtin_amdgcn_wmma_f32_16x16x4_f32(false, a, false, b, (short)0, c,
                                              false, false);
  }

  if (cvalid) {
    int rbase = (lane >> 4) * 8;
#pragma unroll
    for (int r = 0; r < 8; ++r) {
      int mr = row0 + rbase + r;
      if (mr < n) logits[(size_t)mr * 7 + ncol] = c[r];
    }
  }
}

// ---------------------------------------------------------------------------
// per-row log_softmax over 7 classes
// ---------------------------------------------------------------------------
__global__ __launch_bounds__(256) void k_logsoftmax7(const float* __restrict__ logits,
                                                     float* __restrict__ out, int n) {
  int i = blockIdx.x * blockDim.x + threadIdx.x;
  if (i >= n) return;
  const float* r = logits + (size_t)i * 7;
  float v[7];
#pragma unroll
  for (int j = 0; j < 7; ++j) v[j] = r[j];
  float mx = v[0];
#pragma unroll
  for (int j = 1; j < 7; ++j) mx = fmaxf(mx, v[j]);
  float s = 0.0f;
#pragma unroll
  for (int j = 0; j < 7; ++j) s += expf(v[j] - mx);
  float ls = logf(s);
  float* o = out + (size_t)i * 7;
#pragma unroll
  for (int j = 0; j < 7; ++j) o[j] = v[j] - mx - ls;
}

// ---------------------------------------------------------------------------
extern "C" void kernel_launch(void* const* d_in, const int* in_sizes, int n_in,
                              void* d_out, int out_size, void* d_ws, size_t ws_size,
                              hipStream_t stream) {
  (void)n_in; (void)out_size; (void)ws_size;
  const float* x   = (const float*)d_in[0];
  const int*   ei  = (const int*)d_in[1];
  const float* W1l = (const float*)d_in[2];
  const float* b1  = (const float*)d_in[3];
  const float* W1r = (const float*)d_in[4];
  const float* W2l = (const float*)d_in[5];
  const float* b2  = (const float*)d_in[6];
  const float* W2r = (const float*)d_in[7];
  const float* W3l = (const float*)d_in[8];
  const float* b3  = (const float*)d_in[9];
  const float* W3r = (const float*)d_in[10];

  const int n = in_sizes[0] / F;     // 100000 nodes
  const int E = in_sizes[1] / 2;     // 1.6M edges
  const int* src = ei;               // edge_index[0]
  const int* dst = ei + E;           // edge_index[1]

  // workspace layout: deg | agg | hA | hB  (256B aligned regions)
  char* base = (char*)d_ws;
  size_t off = 0;
  auto a256 = [](size_t b) { return (b + 255) & ~(size_t)255; };
  float* deg = (float*)(base + off); off = a256(off + (size_t)n * sizeof(float));
  float* agg = (float*)(base + off); off = a256(off + (size_t)n * F * sizeof(float));
  float* hA  = (float*)(base + off); off = a256(off + (size_t)n * F * sizeof(float));
  float* hB  = (float*)(base + off); off = a256(off + (size_t)n * F * sizeof(float));

  const int T = 256;
  auto nb = [&](long long work) { return (int)((work + T - 1) / T); };
  const long long scatW = (long long)E * 32;      // 32 lanes per edge
  const int gemmB = (n + 127) / 128;              // 8 waves x 16 rows per block

  // in-degree -> 1/deg (mean aggregation)
  k_zero<<<nb(n), T, 0, stream>>>(deg, (size_t)n);
  k_degree<<<nb(E), T, 0, stream>>>(dst, deg, E);
  k_deg_inv<<<nb(n), T, 0, stream>>>(deg, n);

  // layer 1: agg = sum_{e} x[src] at dst ; hA = relu(sage(agg, x))
  k_zero<<<nb((long long)n * F), T, 0, stream>>>(agg, (size_t)n * F);
  k_scatter<<<nb(scatW), T, 0, stream>>>(src, dst, x, agg, E);
  k_gemm128<<<gemmB, T, 0, stream>>>(agg, x, W1l, b1, W1r, deg, hA, n, 1);

  // layer 2
  k_zero<<<nb((long long)n * F), T, 0, stream>>>(agg, (size_t)n * F);
  k_scatter<<<nb(scatW), T, 0, stream>>>(src, dst, hA, agg, E);
  k_gemm128<<<gemmB, T, 0, stream>>>(agg, hA, W2l, b2, W2r, deg, hB, n, 1);

  // layer 3 (logits into hA region, now free) + log_softmax
  k_zero<<<nb((long long)n * F), T, 0, stream>>>(agg, (size_t)n * F);
  k_scatter<<<nb(scatW), T, 0, stream>>>(src, dst, hB, agg, E);
  k_gemm_out7<<<gemmB, T, 0, stream>>>(agg, hB, W3l, b3, W3r, deg, hA, n);

  k_logsoftmax7<<<nb(n), T, 0, stream>>>(hA, (float*)d_out, n);
}